// MambaDynamics_43224550866990
// MI455X (gfx1250) — compile-verified
//
#include <hip/hip_runtime.h>
#include <math.h>

// Problem constants (fixed by the reference)
#define B_      4
#define T_      4096
#define D_      256
#define INNER_  512
#define DSTATE_ 16
#define M_      (B_ * T_)        // 16384 rows for all GEMMs
#define NCHUNK  64               // chunks for the parallel scan
#define CHUNK   (T_ / NCHUNK)    // 64 timesteps per chunk

typedef __attribute__((ext_vector_type(2))) float        v2f;
typedef __attribute__((ext_vector_type(8))) float        v8f;
typedef __attribute__((ext_vector_type(4))) unsigned int u32x4;
typedef __attribute__((ext_vector_type(4))) int          i32x4;
typedef __attribute__((ext_vector_type(8))) int          i32x8;

// TDM builtin arity differs between toolchains; the 6-arg flavor ships the
// gfx1250 TDM header, key off that.
#if defined(__has_include)
#if __has_include(<hip/amd_detail/amd_gfx1250_TDM.h>)
#define TDM_6ARG 1
#else
#define TDM_6ARG 0
#endif
#else
#define TDM_6ARG 0
#endif

enum { ACT_NONE = 0, ACT_SILU = 1, ACT_SOFTPLUS = 2 };

__device__ __forceinline__ float act_apply(int mode, float v) {
    if (mode == ACT_SILU)     return v / (1.f + __expf(-v));
    if (mode == ACT_SOFTPLUS) return (v > 20.f) ? v : log1pf(__expf(v));
    return v;
}

// ---------------------------------------------------------------------------
// Tensor Data Mover: issue a 2D tile load (tile1 rows x tile0 f32 elems) from
// a row-major global tensor (row stride stride0 elems) into LDS at lds_byte.
// pad_enable + pad_interval/pad_amount make the TDM insert one pad DWORD after
// every 2^(padcode+1) DWORDs -> LDS row stride tile0+1 (bank-conflict-free
// fragment reads) with zero VALU cost. D# packing per CDNA5 ISA §8.3/8.4.
// ---------------------------------------------------------------------------
__device__ __forceinline__ void tdm_load_tile_f32(const float* gsrc, unsigned lds_byte,
                                                  unsigned tdim0, unsigned tdim1,
                                                  unsigned tile0, unsigned tile1,
                                                  unsigned stride0, unsigned padcode)
{
    const unsigned long long ga = (unsigned long long)(uintptr_t)gsrc;
    u32x4 g0;
    g0.x = 1u;                                            // count=1, user mode, no gather
    g0.y = lds_byte;                                      // lds_addr
    g0.z = (unsigned)(ga & 0xFFFFFFFFu);                  // global_addr[31:0]
    g0.w = (unsigned)((ga >> 32) & 0x01FFFFFFu) | (2u << 30); // global_addr[56:32], type=2

    i32x8 g1;
    g1[0] = (int)((2u << 16) | (1u << 20) | (padcode << 22)); // data_size=4B, pad_enable,
                                                              // pad_interval=code, pad_amount=0(->1 DW)
    g1[1] = (int)((tdim0 & 0xFFFFu) << 16);               // [63:48] tensor_dim0 lo16 (barrier addr 0)
    g1[2] = (int)(((tdim0 >> 16) & 0xFFFFu) | ((tdim1 & 0xFFFFu) << 16)); // dim0 hi16 | dim1 lo16
    g1[3] = (int)(((tdim1 >> 16) & 0xFFFFu) | ((tile0 & 0xFFFFu) << 16)); // dim1 hi16 | tile_dim0
    g1[4] = (int)(tile1 & 0xFFFFu);                       // tile_dim1 (tile_dim2=0)
    g1[5] = (int)(stride0);                               // tensor_dim0_stride[31:0]
    g1[6] = 0;                                            // stride hi16 | dim1_stride lo16
    g1[7] = 0;                                            // dim1_stride hi32
    i32x4 z4 = {0, 0, 0, 0};
#if TDM_6ARG
    i32x8 z8 = {0, 0, 0, 0, 0, 0, 0, 0};
    __builtin_amdgcn_tensor_load_to_lds(g0, g1, z4, z4, z8, 0);
#else
    __builtin_amdgcn_tensor_load_to_lds(g0, g1, z4, z4, 0);
#endif
}

// ---------------------------------------------------------------------------
// Tiled fp32 WMMA GEMM:  O[M,N] = act(A[M,K](lda) @ W[K,N] + bias[N])
// Block tile 64(M) x 32(N), 256 threads = 8 waves, each wave owns one 16x16
// accumulator. K staged 16 at a time into a double-buffered LDS tile filled by
// the Tensor Data Mover (wave 0 issues, s_wait_tensorcnt + barrier syncs).
// ---------------------------------------------------------------------------
template <int ACT>
__global__ __launch_bounds__(256)
void gemm_wmma(const float* __restrict__ Ag, int lda,
               const float* __restrict__ Wg,
               const float* __restrict__ bias,
               float* __restrict__ Og,
               int M, int N, int K)
{
    // TDM writes rows padded to +1 DWORD: A rows stride 17, W rows stride 33
    __shared__ float sA[2][64][17];
    __shared__ float sB[2][16][33];

    const int tid  = threadIdx.x;
    const int lane = tid & 31;
    const int wave = tid >> 5;
    const int wm   = wave & 3;      // 4 row tiles of 16
    const int wn   = wave >> 2;     // 2 col tiles of 16
    const int half = lane >> 4;     // lane 16..31 selector
    const int l15  = lane & 15;

    const int rowBase = blockIdx.y * 64;
    const int colBase = blockIdx.x * 32;

    v8f acc = {0.f, 0.f, 0.f, 0.f, 0.f, 0.f, 0.f, 0.f};

    if (wave == 0) {   // prologue: fill buffer 0 via TDM
        tdm_load_tile_f32(Ag + (size_t)rowBase * lda,
                          (unsigned)(uintptr_t)&sA[0][0][0],
                          (unsigned)lda, (unsigned)M, 16u, 64u, (unsigned)lda, 3u);
        tdm_load_tile_f32(Wg + colBase,
                          (unsigned)(uintptr_t)&sB[0][0][0],
                          (unsigned)N, (unsigned)K, 32u, 16u, (unsigned)N, 4u);
    }

    for (int kb = 0; kb < K; kb += 16) {
        const int cur = (kb >> 4) & 1;
        if (wave == 0) {
            if (kb + 16 < K) {   // issue next pair, then wait for current pair
                tdm_load_tile_f32(Ag + (size_t)rowBase * lda + (kb + 16),
                                  (unsigned)(uintptr_t)&sA[cur ^ 1][0][0],
                                  (unsigned)lda, (unsigned)M, 16u, 64u, (unsigned)lda, 3u);
                tdm_load_tile_f32(Wg + (size_t)(kb + 16) * N + colBase,
                                  (unsigned)(uintptr_t)&sB[cur ^ 1][0][0],
                                  (unsigned)N, (unsigned)K, 32u, 16u, (unsigned)N, 4u);
                __builtin_amdgcn_s_wait_tensorcnt(2);
            } else {
                __builtin_amdgcn_s_wait_tensorcnt(0);
            }
        }
        __syncthreads();

#pragma unroll
        for (int kk = 0; kk < 4; ++kk) {
            // A 16x4 fragment: lanes 0-15 -> K0(v0),K1(v1); lanes 16-31 -> K2,K3
            const int k0 = kk * 4 + (half ? 2 : 0);
            const int k1 = kk * 4 + (half ? 3 : 1);
            v2f a, b;
            a.x = sA[cur][wm * 16 + l15][k0];
            a.y = sA[cur][wm * 16 + l15][k1];
            // B 4x16 fragment: rows striped across lanes within a VGPR
            b.x = sB[cur][k0][wn * 16 + l15];
            b.y = sB[cur][k1][wn * 16 + l15];
            acc = __builtin_amdgcn_wmma_f32_16x16x4_f32(
                false, a, false, b, (short)0, acc, false, false);
        }
        __syncthreads();
    }

    // C/D layout: VGPR i -> row i (lanes 0-15) / row i+8 (lanes 16-31), col = lane&15
    const int col = colBase + wn * 16 + l15;
    const float bs = bias[col];
#pragma unroll
    for (int i = 0; i < 8; ++i) {
        const int row = rowBase + wm * 16 + (half ? 8 : 0) + i;
        float v = acc[i] + bs;
        Og[(size_t)row * N + col] = act_apply(ACT, v);
    }
}

// ---------------------------------------------------------------------------
// A = -exp(A_log)
// ---------------------------------------------------------------------------
__global__ void aprep_kernel(const float* __restrict__ Alog, float* __restrict__ Aneg)
{
    int i = blockIdx.x * blockDim.x + threadIdx.x;
    if (i < INNER_ * DSTATE_) Aneg[i] = -__expf(Alog[i]);
}

// ---------------------------------------------------------------------------
// Chunked scan, pass A: per chunk compute P = prod(A_bar) and H = zero-start
// partial state, for every (b, ch, s).
// ---------------------------------------------------------------------------
__global__ __launch_bounds__(256)
void scan_partial(const float* __restrict__ xconv, const float* __restrict__ dt,
                  const float* __restrict__ bc, const float* __restrict__ Aneg,
                  float* __restrict__ Pbuf, float* __restrict__ Hbuf)
{
    const int ch = blockIdx.x * blockDim.x + threadIdx.x;  // 0..INNER-1
    const int c  = blockIdx.y;                             // chunk
    const int b  = blockIdx.z;                             // batch

    float Ar[DSTATE_], h[DSTATE_], P[DSTATE_];
#pragma unroll
    for (int s = 0; s < DSTATE_; ++s) {
        Ar[s] = Aneg[ch * DSTATE_ + s];
        h[s] = 0.f;
        P[s] = 1.f;
    }

    const int t0 = c * CHUNK;
    for (int i = 0; i < CHUNK; ++i) {
        const size_t rt = (size_t)(b * T_ + t0 + i);
        const float x  = xconv[rt * INNER_ + ch];
        const float d  = dt[rt * INNER_ + ch];
        const float dx = d * x;
        const float* bcp = &bc[rt * 32];   // uniform address -> scalar loads
#pragma unroll
        for (int s = 0; s < DSTATE_; ++s) {
            const float ab = __expf(d * Ar[s]);
            P[s] *= ab;
            h[s] = ab * h[s] + dx * bcp[s];
        }
    }

    float* pp = &Pbuf[(((size_t)b * NCHUNK + c) * INNER_ + ch) * DSTATE_];
    float* hp = &Hbuf[(((size_t)b * NCHUNK + c) * INNER_ + ch) * DSTATE_];
#pragma unroll
    for (int s = 0; s < DSTATE_; ++s) { pp[s] = P[s]; hp[s] = h[s]; }
}

// ---------------------------------------------------------------------------
// Pass B: tiny serial combine across the 64 chunk carries; parallel across all
// B*INNER*DSTATE = 32768 state chains. Writes the initial h for each chunk.
// ---------------------------------------------------------------------------
__global__ __launch_bounds__(256)
void scan_carry(const float* __restrict__ Pbuf, const float* __restrict__ Hbuf,
                float* __restrict__ hinit)
{
    const int gid = blockIdx.x * blockDim.x + threadIdx.x;  // [B][INNER*DSTATE]
    const int b   = gid / (INNER_ * DSTATE_);
    const int cs  = gid % (INNER_ * DSTATE_);
    float h = 0.f;
    for (int c = 0; c < NCHUNK; ++c) {
        const size_t off = ((size_t)b * NCHUNK + c) * (INNER_ * DSTATE_) + cs;
        hinit[off] = h;
        h = Pbuf[off] * h + Hbuf[off];
    }
}

// ---------------------------------------------------------------------------
// Pass C: replay each chunk from its correct initial state, emit
// y[b,t,ch] = (sum_s h*C) * silu(gate).
// ---------------------------------------------------------------------------
__global__ __launch_bounds__(256)
void scan_final(const float* __restrict__ xconv, const float* __restrict__ dt,
                const float* __restrict__ bc, const float* __restrict__ Aneg,
                const float* __restrict__ hinit, const float* __restrict__ xproj,
                float* __restrict__ ybuf)
{
    const int ch = blockIdx.x * blockDim.x + threadIdx.x;
    const int c  = blockIdx.y;
    const int b  = blockIdx.z;

    float Ar[DSTATE_], h[DSTATE_];
    const float* hp = &hinit[(((size_t)b * NCHUNK + c) * INNER_ + ch) * DSTATE_];
#pragma unroll
    for (int s = 0; s < DSTATE_; ++s) {
        Ar[s] = Aneg[ch * DSTATE_ + s];
        h[s] = hp[s];
    }

    const int t0 = c * CHUNK;
    for (int i = 0; i < CHUNK; ++i) {
        const size_t rt = (size_t)(b * T_ + t0 + i);
        const float x  = xconv[rt * INNER_ + ch];
        const float d  = dt[rt * INNER_ + ch];
        const float dx = d * x;
        const float* bcp = &bc[rt * 32];
        float y = 0.f;
#pragma unroll
        for (int s = 0; s < DSTATE_; ++s) {
            const float ab = __expf(d * Ar[s]);
            h[s] = ab * h[s] + dx * bcp[s];          // B_proj = bcp[0..15]
            y += h[s] * bcp[16 + s];                 // C_proj = bcp[16..31]
        }
        const float g = xproj[rt * 1024 + 512 + ch]; // x_gate half of x_proj
        ybuf[rt * INNER_ + ch] = y * (g / (1.f + __expf(-g)));
    }
}

// ---------------------------------------------------------------------------
// Launch: 4 TDM+WMMA GEMMs + A-prep + 3-pass chunked scan + output GEMM.
// ---------------------------------------------------------------------------
extern "C" void kernel_launch(void* const* d_in, const int* in_sizes, int n_in,
                              void* d_out, int out_size, void* d_ws, size_t ws_size,
                              hipStream_t stream)
{
    const float* x      = (const float*)d_in[0];
    const float* W_in   = (const float*)d_in[1];
    const float* b_in   = (const float*)d_in[2];
    const float* W_conv = (const float*)d_in[3];
    const float* b_conv = (const float*)d_in[4];
    const float* A_log  = (const float*)d_in[5];
    const float* W_bc   = (const float*)d_in[6];
    const float* b_bc   = (const float*)d_in[7];
    const float* W_dt   = (const float*)d_in[8];
    const float* b_dt   = (const float*)d_in[9];
    const float* W_out  = (const float*)d_in[10];
    const float* b_out  = (const float*)d_in[11];
    float* out = (float*)d_out;

    float* ws = (float*)d_ws;
    size_t o = 0;
    float* xproj = ws + o; o += (size_t)M_ * 1024;                     // [M,2*INNER]
    float* xconv = ws + o; o += (size_t)M_ * INNER_;                   // [M,INNER]
    float* dtb   = ws + o; o += (size_t)M_ * INNER_;                   // [M,INNER]
    float* bcb   = ws + o; o += (size_t)M_ * 32;                       // [M,2*DSTATE]
    float* ybuf  = ws + o; o += (size_t)M_ * INNER_;                   // [M,INNER]
    float* Aneg  = ws + o; o += (size_t)INNER_ * DSTATE_;
    float* Pbuf  = ws + o; o += (size_t)B_ * NCHUNK * INNER_ * DSTATE_;
    float* Hbuf  = ws + o; o += (size_t)B_ * NCHUNK * INNER_ * DSTATE_;
    float* hinit = ws + o; o += (size_t)B_ * NCHUNK * INNER_ * DSTATE_;

    const dim3 blk(256);

    // x_proj = x @ W_in + b_in             [16384,256]@[256,1024]
    gemm_wmma<ACT_NONE><<<dim3(1024 / 32, M_ / 64), blk, 0, stream>>>(
        x, D_, W_in, b_in, xproj, M_, 1024, D_);
    // x_conv = silu(x_main @ W_conv + b)   [16384,512(lda 1024)]@[512,512]
    gemm_wmma<ACT_SILU><<<dim3(512 / 32, M_ / 64), blk, 0, stream>>>(
        xproj, 1024, W_conv, b_conv, xconv, M_, 512, 512);
    // x_bc = x_conv @ W_bc + b_bc          [16384,512]@[512,32]
    gemm_wmma<ACT_NONE><<<dim3(32 / 32, M_ / 64), blk, 0, stream>>>(
        xconv, 512, W_bc, b_bc, bcb, M_, 32, 512);
    // dt = softplus(x_conv @ W_dt + b_dt)  [16384,512]@[512,512]
    gemm_wmma<ACT_SOFTPLUS><<<dim3(512 / 32, M_ / 64), blk, 0, stream>>>(
        xconv, 512, W_dt, b_dt, dtb, M_, 512, 512);

    aprep_kernel<<<(INNER_ * DSTATE_ + 255) / 256, blk, 0, stream>>>(A_log, Aneg);

    scan_partial<<<dim3(INNER_ / 256, NCHUNK, B_), blk, 0, stream>>>(
        xconv, dtb, bcb, Aneg, Pbuf, Hbuf);
    scan_carry<<<(B_ * INNER_ * DSTATE_) / 256, blk, 0, stream>>>(Pbuf, Hbuf, hinit);
    scan_final<<<dim3(INNER_ / 256, NCHUNK, B_), blk, 0, stream>>>(
        xconv, dtb, bcb, Aneg, hinit, xproj, ybuf);

    // out = y @ W_out + b_out              [16384,512]@[512,256]
    gemm_wmma<ACT_NONE><<<dim3(256 / 32, M_ / 64), blk, 0, stream>>>(
        ybuf, 512, W_out, b_out, out, M_, 256, 512);

    (void)in_sizes; (void)n_in; (void)out_size; (void)ws_size;
}